// DiscriminativeLoss_63634235458250
// MI455X (gfx1250) — compile-verified
//
#include <hip/hip_runtime.h>

typedef __attribute__((ext_vector_type(2))) float v2f;
typedef __attribute__((ext_vector_type(8))) float v8f;

#define HW   (512 * 512)
#define NB   16
#define NF   16
#define NK   8
#define NBLK 128
#define TPB  256

// workspace layout (float offsets)
#define WS_SEG  0      // [16][8][16]
#define WS_CNT  2048   // [16][8]
#define WS_VAR  2176   // [16][8]
#define WS_DIST 2304   // [16]
#define WS_REG  2320   // [16]
#define WS_MU   2336   // [16][8][16]  (not atomically accumulated)
#define WS_ZERO_FLOATS 2336

// ---------------------------------------------------------------------------
// Phase 1: segment-sum via one-hot WMMA (D = onehot^T * X), plus label counts.
// acc layout (f32 16x16 C/D): element k, lanes 0-15 -> seg[label k][feature lane]
// ---------------------------------------------------------------------------
__global__ __launch_bounds__(TPB) void seg_sum_kernel(const float* __restrict__ pred,
                                                      const int* __restrict__ tgt,
                                                      float* __restrict__ ws) {
    __shared__ float xs[8][32 * 17];  // per-wave transpose tile, 17-padded
    __shared__ int   ls[8][32];       // per-wave labels

    const int tid  = threadIdx.x;
    const int lane = tid & 31;
    const int w    = tid >> 5;
    const int b    = blockIdx.y;
    const int half = lane >> 4;   // 0: K=0,1  1: K=2,3
    const int n    = lane & 15;   // A: label row m ; B: feature col n

    const float* p = pred + (size_t)b * NF * HW;
    const int*   t = tgt  + (size_t)b * HW;

    v8f acc = {};
    float cnt[NK];
#pragma unroll
    for (int k = 0; k < NK; ++k) cnt[k] = 0.0f;

    // HW / (NBLK*TPB) == 8 exactly -> uniform trip count, EXEC all-ones for WMMA
    for (int pix = blockIdx.x * TPB + tid; pix < HW; pix += NBLK * TPB) {
        const int lab = t[pix];
        ls[w][lane] = lab;
#pragma unroll
        for (int f = 0; f < NF; ++f)                 // coalesced plane loads,
            xs[w][lane * 17 + f] = p[(size_t)f * HW + pix];  // conflict-free LDS transpose
#pragma unroll
        for (int k = 0; k < NK; ++k) cnt[k] += (lab == k) ? 1.0f : 0.0f;

        // per-wave private LDS region; DS ops from the same wave are in-order
#pragma unroll
        for (int c = 0; c < 8; ++c) {                // 8 chunks x 4 pixels = 32 pixels
            const int p0 = c * 4 + half * 2;
            const float b0 = xs[w][p0 * 17 + n];
            const float b1 = xs[w][(p0 + 1) * 17 + n];
            const int la0 = ls[w][p0];
            const int la1 = ls[w][p0 + 1];
            v2f A = { (la0 == n) ? 1.0f : 0.0f, (la1 == n) ? 1.0f : 0.0f };
            v2f Bv = { b0, b1 };
            acc = __builtin_amdgcn_wmma_f32_16x16x4_f32(
                false, A, false, Bv, (short)0, acc, false, false);
        }
    }

    // rows 8..15 (lanes 16-31) are identically zero: labels < 8
    if (lane < 16) {
#pragma unroll
        for (int k = 0; k < NK; ++k)
            atomicAdd(&ws[WS_SEG + (b * NK + k) * NF + n], acc[k]);
    }
#pragma unroll
    for (int k = 0; k < NK; ++k) {
        float v = cnt[k];
#pragma unroll
        for (int off = 16; off > 0; off >>= 1) v += __shfl_down(v, off, 32);
        if (lane == 0) atomicAdd(&ws[WS_CNT + b * NK + k], v);
    }
}

// ---------------------------------------------------------------------------
// Phase 2: mu = seg / max(cnt,1); l_reg and pairwise l_dist. One block, 128 thr.
// ---------------------------------------------------------------------------
__global__ __launch_bounds__(128) void finalize_mu_kernel(float* __restrict__ ws) {
    const int t = threadIdx.x;      // (b, k) pair
    const int b = t >> 3;
    const int k = t & 7;

    const float c = fmaxf(ws[WS_CNT + b * NK + k], 1.0f);
    float mu[NF];
    float s = 0.0f;
#pragma unroll
    for (int f = 0; f < NF; ++f) {
        mu[f] = ws[WS_SEG + (b * NK + k) * NF + f] / c;
        ws[WS_MU + (b * NK + k) * NF + f] = mu[f];
        s = fmaf(mu[f], mu[f], s);
    }
    atomicAdd(&ws[WS_REG + b], sqrtf(s));
    __syncthreads();

    float dsum = 0.0f;
    for (int j = 0; j < NK; ++j) {
        if (j == k) continue;
        float a = 0.0f;
#pragma unroll
        for (int f = 0; f < NF; ++f) {
            const float d = mu[f] - ws[WS_MU + (b * NK + j) * NF + f];
            a = fmaf(d, d, a);
        }
        float h = 3.0f - sqrtf(a);   // 2 * DELTA_D - ||mu_k - mu_j||
        h = fmaxf(h, 0.0f);
        dsum = fmaf(h, h, dsum);
    }
    atomicAdd(&ws[WS_DIST + b], dsum);
}

// ---------------------------------------------------------------------------
// Phase 3: second streaming pass: per-pixel hinge pull, segment-summed by label.
// ---------------------------------------------------------------------------
__global__ __launch_bounds__(TPB) void lvar_kernel(const float* __restrict__ pred,
                                                   const int* __restrict__ tgt,
                                                   float* __restrict__ ws) {
    __shared__ float mus[NK * 17];  // padded: gather mu[lab][f] is conflict-free
    const int tid  = threadIdx.x;
    const int lane = tid & 31;
    const int b    = blockIdx.y;

    if (tid < 128) {
        const int k = tid >> 4, f = tid & 15;
        mus[k * 17 + f] = ws[WS_MU + (b * NK + k) * NF + f];
    }
    __syncthreads();

    const float* p = pred + (size_t)b * NF * HW;
    const int*   t = tgt  + (size_t)b * HW;

    float vs[NK];
#pragma unroll
    for (int k = 0; k < NK; ++k) vs[k] = 0.0f;

    for (int pix = blockIdx.x * TPB + tid; pix < HW; pix += NBLK * TPB) {
        const int lab = t[pix];
        float a = 0.0f;
#pragma unroll
        for (int f = 0; f < NF; ++f) {
            const float x = p[(size_t)f * HW + pix];
            const float d = mus[lab * 17 + f] - x;
            a = fmaf(d, d, a);
        }
        float d = sqrtf(a) - 0.5f;   // - DELTA_V
        d = fmaxf(d, 0.0f);
        const float d2 = d * d;
#pragma unroll
        for (int k = 0; k < NK; ++k) vs[k] += (lab == k) ? d2 : 0.0f;
    }

#pragma unroll
    for (int k = 0; k < NK; ++k) {
        float v = vs[k];
#pragma unroll
        for (int off = 16; off > 0; off >>= 1) v += __shfl_down(v, off, 32);
        if (lane == 0) atomicAdd(&ws[WS_VAR + b * NK + k], v);
    }
}

// ---------------------------------------------------------------------------
// Phase 4: per-image combine + mean over B. One wave.
// ---------------------------------------------------------------------------
__global__ __launch_bounds__(32) void reduce_kernel(const float* __restrict__ ws,
                                                    float* __restrict__ out) {
    const int lane = threadIdx.x;
    float lv = 0.0f, ld = 0.0f, lr = 0.0f, lo = 0.0f;
    if (lane < NB) {
        const int b = lane;
        float v = 0.0f;
#pragma unroll
        for (int k = 0; k < NK; ++k)
            v += ws[WS_VAR + b * NK + k] / fmaxf(ws[WS_CNT + b * NK + k], 1.0f);
        v *= (1.0f / (float)NK);
        const float dd = ws[WS_DIST + b] * (1.0f / (float)(NK * (NK - 1)));
        const float rr = ws[WS_REG + b] * (1.0f / (float)NK);
        lv = v; ld = dd; lr = rr;
        lo = v + dd + 0.001f * rr;   // ALPHA, BETA, GAMMA
    }
#pragma unroll
    for (int off = 16; off > 0; off >>= 1) {
        lo += __shfl_down(lo, off, 32);
        lv += __shfl_down(lv, off, 32);
        ld += __shfl_down(ld, off, 32);
        lr += __shfl_down(lr, off, 32);
    }
    if (lane == 0) {
        out[0] = lo * (1.0f / (float)NB);
        out[1] = lv * (1.0f / (float)NB);
        out[2] = ld * (1.0f / (float)NB);
        out[3] = lr * (1.0f / (float)NB);
    }
}

extern "C" void kernel_launch(void* const* d_in, const int* in_sizes, int n_in,
                              void* d_out, int out_size, void* d_ws, size_t ws_size,
                              hipStream_t stream) {
    const float* pred = (const float*)d_in[0];
    const int*   tgt  = (const int*)d_in[1];
    float* ws  = (float*)d_ws;
    float* out = (float*)d_out;

    hipMemsetAsync(d_ws, 0, WS_ZERO_FLOATS * sizeof(float), stream);

    dim3 grid(NBLK, NB);
    seg_sum_kernel<<<grid, TPB, 0, stream>>>(pred, tgt, ws);
    finalize_mu_kernel<<<1, 128, 0, stream>>>(ws);
    lvar_kernel<<<grid, TPB, 0, stream>>>(pred, tgt, ws);
    reduce_kernel<<<1, 32, 0, stream>>>(ws, out);
}